// GCNEncoder_36206574305699
// MI455X (gfx1250) — compile-verified
//
#include <hip/hip_runtime.h>
#include <hip/hip_bf16.h>

// GCN encoder: 3 layers of { xw = h@W^T (WMMA f32) ; scatter-add over edges ;
// +bias ; LayerNorm ; (ReLU) ; +residual }.
// N=50000, D=128, E=1600000, L=3.

#define GCN_N 50000
#define GCN_E 1600000
#define GCN_D 128
#define GCN_L 3
#define LN_EPS 1e-5f

typedef __attribute__((ext_vector_type(2))) float v2f;
typedef __attribute__((ext_vector_type(8))) float v8f;

// ---------------- degree / norm precompute ----------------

__global__ void k_init_deg(float* __restrict__ deg, int n) {
    int i = blockIdx.x * blockDim.x + threadIdx.x;
    if (i < n) deg[i] = 1.0f;  // self-loop weight
}

__global__ void k_accum_deg(const float* __restrict__ w, const int* __restrict__ col,
                            float* __restrict__ deg, int e) {
    int i = blockIdx.x * blockDim.x + threadIdx.x;
    if (i < e) atomicAdd(&deg[col[i]], w[i]);
}

__global__ void k_dinv(const float* __restrict__ deg, float* __restrict__ dinv, int n) {
    int i = blockIdx.x * blockDim.x + threadIdx.x;
    if (i < n) {
        float d = deg[i];
        dinv[i] = (d > 0.0f) ? rsqrtf(fmaxf(d, 1e-12f)) : 0.0f;
    }
}

__global__ void k_norm(const float* __restrict__ dinv, const int* __restrict__ row,
                       const int* __restrict__ col, const float* __restrict__ w,
                       float* __restrict__ norm, int e) {
    int i = blockIdx.x * blockDim.x + threadIdx.x;
    if (i < e) norm[i] = dinv[row[i]] * w[i] * dinv[col[i]];
}

// ---------------- GEMM: XW[n][j] = sum_k H[n][k] * W[j][k] ----------------
// One wave per 16x16 output tile; K-loop of V_WMMA_F32_16X16X4_F32.
// A (16x4 f32) lane layout: lanes 0-15 M=lane {K=k,k+1}; lanes 16-31 M=lane-16 {K=k+2,k+3}.
// B (4x16 f32): lanes 0-15 N=lane {K=k,k+1}; lanes 16-31 N=lane-16 {K=k+2,k+3}.
// C/D (16x16 f32): VGPR v -> M=v (lanes 0-15, N=lane) / M=v+8 (lanes 16-31, N=lane-16).

__global__ void k_gemm_wmma(const float* __restrict__ H, const float* __restrict__ W,
                            float* __restrict__ XW, int n_nodes) {
    const int lane = threadIdx.x & 31;
    const int wave_in_block = threadIdx.x >> 5;
    const int waves_per_block = blockDim.x >> 5;
    const int tile_id = blockIdx.x * waves_per_block + wave_in_block;
    const int tiles_n = GCN_D / 16;                 // 8 column tiles
    const int tile_m = tile_id / tiles_n;
    const int tile_n = tile_id % tiles_n;
    if (tile_m * 16 >= n_nodes) return;             // wave-uniform: EXEC stays all-1s

    const int row0 = tile_m * 16;
    const int col0 = tile_n * 16;
    const int lm = lane & 15;
    const int hi = lane >> 4;                       // 0: K pair {0,1}, 1: K pair {2,3}

    const float2* __restrict__ Ap = (const float2*)(H + (size_t)(row0 + lm) * GCN_D);
    const float2* __restrict__ Bp = (const float2*)(W + (size_t)(col0 + lm) * GCN_D);

    v8f c = {};
#pragma unroll
    for (int k = 0; k < GCN_D; k += 4) {
        const int kp = (k >> 1) + hi;               // float2 index: k/2 (+1 for hi lanes)
        float2 af = Ap[kp];
        float2 bf = Bp[kp];
        v2f a; a.x = af.x; a.y = af.y;
        v2f b; b.x = bf.x; b.y = bf.y;
        c = __builtin_amdgcn_wmma_f32_16x16x4_f32(
                /*neg_a=*/false, a, /*neg_b=*/false, b,
                /*c_mod=*/(short)0, c, /*reuse_a=*/false, /*reuse_b=*/false);
    }

    const int m0 = row0 + hi * 8;
    const int nc = col0 + lm;
#pragma unroll
    for (int v = 0; v < 8; ++v)
        XW[(size_t)(m0 + v) * GCN_D + nc] = c[v];
}

// ---------------- acc init: bias + self-loop contribution ----------------
// acc[n][d] = bias[d] + XW[n][d] * dinv[n]^2   (self-loop norm = dinv*1*dinv)

__global__ void k_init_acc(const float* __restrict__ XW, const float* __restrict__ dinv,
                           const float* __restrict__ bias, float* __restrict__ acc,
                           int n_nodes) {
    const int wave = (blockIdx.x * blockDim.x + threadIdx.x) >> 5;
    const int lane = threadIdx.x & 31;
    if (wave >= n_nodes) return;
    const float self = dinv[wave] * dinv[wave];
    float4 x = ((const float4*)(XW + (size_t)wave * GCN_D))[lane];
    float4 b = ((const float4*)bias)[lane];
    float4 o;
    o.x = b.x + x.x * self;
    o.y = b.y + x.y * self;
    o.z = b.z + x.z * self;
    o.w = b.w + x.w * self;
    ((float4*)(acc + (size_t)wave * GCN_D))[lane] = o;
}

// ---------------- edge scatter: acc[col] += XW[row] * norm ----------------
// One wave per edge; each lane moves a float4 (32 lanes * 4 = 128 features).
// XW (25.6 MB) is L2-resident on MI455X (192 MB L2) -> gathers + atomics hit L2.

__global__ void k_scatter(const float* __restrict__ XW, const int* __restrict__ row,
                          const int* __restrict__ col, const float* __restrict__ norm,
                          float* __restrict__ acc, int n_edges) {
    const int wave = (blockIdx.x * blockDim.x + threadIdx.x) >> 5;
    const int lane = threadIdx.x & 31;
    if (wave >= n_edges) return;
    const int r = row[wave];
    const int c = col[wave];
    const float w = norm[wave];
    float4 m = ((const float4*)(XW + (size_t)r * GCN_D))[lane];
    float* dst = acc + (size_t)c * GCN_D + lane * 4;
    atomicAdd(dst + 0, m.x * w);
    atomicAdd(dst + 1, m.y * w);
    atomicAdd(dst + 2, m.z * w);
    atomicAdd(dst + 3, m.w * w);
}

// ---------------- LayerNorm + (ReLU) + residual ----------------
// One wave per node; wave32 shfl_xor reductions for mean/var over 128 features.

__global__ void k_ln_res(const float* __restrict__ acc, const float* __restrict__ hin,
                         const float* __restrict__ gamma, const float* __restrict__ beta,
                         float* __restrict__ hout, int n_nodes, int do_relu) {
    const int wave = (blockIdx.x * blockDim.x + threadIdx.x) >> 5;
    const int lane = threadIdx.x & 31;
    if (wave >= n_nodes) return;

    float4 a = ((const float4*)(acc + (size_t)wave * GCN_D))[lane];

    float s = a.x + a.y + a.z + a.w;
#pragma unroll
    for (int off = 16; off > 0; off >>= 1) s += __shfl_xor(s, off, 32);
    const float mu = s * (1.0f / GCN_D);

    float dx = a.x - mu, dy = a.y - mu, dz = a.z - mu, dw = a.w - mu;
    float q = dx * dx + dy * dy + dz * dz + dw * dw;
#pragma unroll
    for (int off = 16; off > 0; off >>= 1) q += __shfl_xor(q, off, 32);
    const float inv = rsqrtf(q * (1.0f / GCN_D) + LN_EPS);

    float4 g = ((const float4*)gamma)[lane];
    float4 b = ((const float4*)beta)[lane];
    float4 id = ((const float4*)(hin + (size_t)wave * GCN_D))[lane];

    float4 y;
    y.x = dx * inv * g.x + b.x;
    y.y = dy * inv * g.y + b.y;
    y.z = dz * inv * g.z + b.z;
    y.w = dw * inv * g.w + b.w;
    if (do_relu) {
        y.x = fmaxf(y.x, 0.0f);
        y.y = fmaxf(y.y, 0.0f);
        y.z = fmaxf(y.z, 0.0f);
        y.w = fmaxf(y.w, 0.0f);
    }
    y.x += id.x; y.y += id.y; y.z += id.z; y.w += id.w;
    ((float4*)(hout + (size_t)wave * GCN_D))[lane] = y;
}

// ---------------- host orchestration ----------------

extern "C" void kernel_launch(void* const* d_in, const int* in_sizes, int n_in,
                              void* d_out, int out_size, void* d_ws, size_t ws_size,
                              hipStream_t stream) {
    (void)in_sizes; (void)n_in; (void)out_size; (void)ws_size;

    const float* x      = (const float*)d_in[0];           // [N, D]
    const int*   ei     = (const int*)d_in[1];             // [2, E] row-major
    const float* ew     = (const float*)d_in[2];           // [E]
    const float* Ws     = (const float*)d_in[3];           // [L, D, D]
    const float* bs     = (const float*)d_in[4];           // [L, D]
    const float* gammas = (const float*)d_in[5];           // [L, D]
    const float* betas  = (const float*)d_in[6];           // [L, D]
    float*       out    = (float*)d_out;                   // [N, D]

    const int* row = ei;            // edge_index[0]
    const int* col = ei + GCN_E;    // edge_index[1]

    // workspace layout (floats)
    float* ws   = (float*)d_ws;
    float* deg  = ws;                                   // N
    float* dinv = deg + GCN_N;                          // N
    float* norm = dinv + GCN_N;                         // E
    float* xw   = norm + GCN_E;                         // N*D
    float* acc  = xw + (size_t)GCN_N * GCN_D;           // N*D
    float* h1   = acc + (size_t)GCN_N * GCN_D;          // N*D

    const int BLK = 256;

    // ---- normalization coefficients (graph is identical every layer) ----
    k_init_deg<<<(GCN_N + BLK - 1) / BLK, BLK, 0, stream>>>(deg, GCN_N);
    k_accum_deg<<<(GCN_E + BLK - 1) / BLK, BLK, 0, stream>>>(ew, col, deg, GCN_E);
    k_dinv<<<(GCN_N + BLK - 1) / BLK, BLK, 0, stream>>>(deg, dinv, GCN_N);
    k_norm<<<(GCN_E + BLK - 1) / BLK, BLK, 0, stream>>>(dinv, row, col, ew, norm, GCN_E);

    // grid sizing
    const int tiles_total  = ((GCN_N + 15) / 16) * (GCN_D / 16);     // 25000 waves
    const int gemm_blocks  = (tiles_total + (BLK / 32) - 1) / (BLK / 32);
    const int node_wave_blocks = (GCN_N + (BLK / 32) - 1) / (BLK / 32);
    const int edge_wave_blocks = (GCN_E + (BLK / 32) - 1) / (BLK / 32);

    // ping-pong: x -> h1 -> out -> out (in-place LN safe: each wave owns its row)
    const float* hin[GCN_L]  = { x,  h1,  out };
    float*       hout[GCN_L] = { h1, out, out };

    for (int l = 0; l < GCN_L; ++l) {
        const float* W = Ws + (size_t)l * GCN_D * GCN_D;
        k_gemm_wmma<<<gemm_blocks, BLK, 0, stream>>>(hin[l], W, xw, GCN_N);
        k_init_acc<<<node_wave_blocks, BLK, 0, stream>>>(xw, dinv, bs + (size_t)l * GCN_D,
                                                         acc, GCN_N);
        k_scatter<<<edge_wave_blocks, BLK, 0, stream>>>(xw, row, col, norm, acc, GCN_E);
        k_ln_res<<<node_wave_blocks, BLK, 0, stream>>>(acc, hin[l],
                                                       gammas + (size_t)l * GCN_D,
                                                       betas + (size_t)l * GCN_D,
                                                       hout[l], GCN_N,
                                                       /*do_relu=*/(l < GCN_L - 1) ? 1 : 0);
    }
}